// GraphConvGRU_10763188044361
// MI455X (gfx1250) — compile-verified
//
#include <hip/hip_runtime.h>
#include <math.h>

#define N_NODES 10000
#define SEQ_LEN 50
#define B_SZ    2
#define H_DIM   64
#define IN_DIM  128
#define E_EDGES 160000

typedef __attribute__((ext_vector_type(2))) float v2f;
typedef __attribute__((ext_vector_type(8))) float v8f;

// workspace layout (float offsets)
#define WS_AGG    0                       // B*N*H = 1,280,000 floats
#define WS_NORMS  1280000                 // 10,000 (deg_out -> norm_s in place)
#define WS_NORMD  1290000                 // 10,000 (deg_in  -> norm_d in place)
#define WS_COEFF  1300000                 // 160,000
#define WS_XPROJ  1460000                 // 3 * B * H = 384 (xr, xz, xh)

// ---------------------------------------------------------------- utilities
__global__ void k_zero(float* __restrict__ p, int n) {
    int i = blockIdx.x * blockDim.x + threadIdx.x;
    if (i < n) p[i] = 0.0f;
}

__global__ void k_degree(const int* __restrict__ src, const int* __restrict__ dst,
                         float* __restrict__ deg_out, float* __restrict__ deg_in) {
    int e = blockIdx.x * blockDim.x + threadIdx.x;
    if (e < E_EDGES) {
        atomicAdd(&deg_out[src[e]], 1.0f);
        atomicAdd(&deg_in[dst[e]], 1.0f);
    }
}

__global__ void k_norm(float* __restrict__ p, int n) {
    int i = blockIdx.x * blockDim.x + threadIdx.x;
    if (i < n) p[i] = rsqrtf(fmaxf(p[i], 1.0f));   // clip(deg,1)^-0.5
}

__global__ void k_coeff(const int* __restrict__ src, const int* __restrict__ dst,
                        const float* __restrict__ ns, const float* __restrict__ nd,
                        float* __restrict__ coeff) {
    int e = blockIdx.x * blockDim.x + threadIdx.x;
    if (e < E_EDGES) coeff[e] = ns[src[e]] * nd[dst[e]];
}

// time-invariant input projections: xr/xz/xh [B,H]
__global__ void k_xproj(const float* __restrict__ x,
                        const float* __restrict__ wr, const float* __restrict__ br,
                        const float* __restrict__ wz, const float* __restrict__ bz,
                        const float* __restrict__ wh, const float* __restrict__ bh,
                        float* __restrict__ xproj) {
    int t = threadIdx.x;              // 0..127 = b*64 + col
    int b = t >> 6, c = t & 63;
    float ar = 0.f, az = 0.f, ah = 0.f;
    for (int k = 0; k < IN_DIM; ++k) {
        float xv = x[b * IN_DIM + k];
        ar += xv * wr[k * H_DIM + c];
        az += xv * wz[k * H_DIM + c];
        ah += xv * wh[k * H_DIM + c];
    }
    xproj[0 * 128 + t] = ar + br[c];
    xproj[1 * 128 + t] = az + bz[c];
    xproj[2 * 128 + t] = ah + bh[c];
}

// ------------------------------------------------------ edge scatter (GCN agg)
// agg[b][dst][:] += h_prev[b][src][:] * coeff[e], h_prev lives in d_out[:,t-1]
__global__ void __launch_bounds__(256)
k_scatter(const float* __restrict__ out, int tprev,
          const int* __restrict__ src, const int* __restrict__ dst,
          const float* __restrict__ coeff, float* __restrict__ agg) {
    int gid = blockIdx.x * blockDim.x + threadIdx.x;
    int e = gid >> 5;                     // one wave32 per edge (2 batches x 64)
    if (e >= E_EDGES) return;
    int lane = threadIdx.x & 31;
    int s = src[e], d = dst[e];
    float c = coeff[e];
    int b   = lane >> 4;                  // lanes 0-15: batch0, 16-31: batch1
    int col = (lane & 15) * 4;
    const float4 hv = *(const float4*)(out +
        (((size_t)(b * SEQ_LEN + tprev)) * N_NODES + (size_t)s) * H_DIM + col);
    float* ap = agg + ((size_t)b * N_NODES + (size_t)d) * H_DIM + col;
    atomicAdd(ap + 0, hv.x * c);
    atomicAdd(ap + 1, hv.y * c);
    atomicAdd(ap + 2, hv.z * c);
    atomicAdd(ap + 3, hv.w * c);
}

// --------------------------------------------- fused GEMM (WMMA f32) + GRU gates
// one wave per 16-row stripe: g = agg @ gcn_w + gcn_b, then GRU update, and
// cooperative re-zero of the stripe's agg tile for the next timestep.
__global__ void __launch_bounds__(256)
k_update(float* __restrict__ agg,
         const float* __restrict__ gcn_w, const float* __restrict__ gcn_b,
         const float* __restrict__ xproj,
         float* __restrict__ out, int tstep) {
    // gcn_w staged pre-swizzled into per-lane v2f B-fragments:
    //   entry e = (kk*4 + t4)*32 + lane  ->  { W[k0][ncol], W[k0+1][ncol] }
    //   with k0 = kk*4 + (lane>>4)*2, ncol = t4*16 + (lane&15)
    // -> inner loop becomes one bank-conflict-free ds_load_b64 per WMMA.
    __shared__ __align__(16) float lds_wswz[2 * 2048];   // 16 KB
    __shared__ float lds_bias[H_DIM];
    __shared__ float lds_x[3 * B_SZ * H_DIM];            // xr|xz|xh

    int tid = threadIdx.x;
#pragma unroll
    for (int ii = 0; ii < 8; ++ii) {
        int i    = tid + 256 * ii;        // fragment-pair index 0..2047
        int kk   = i >> 7;
        int rem  = i & 127;
        int t4   = rem >> 5;
        int ln   = rem & 31;
        int k0   = kk * 4 + (ln >> 4) * 2;
        int ncol = t4 * 16 + (ln & 15);
        lds_wswz[2 * i + 0] = gcn_w[k0 * H_DIM + ncol];
        lds_wswz[2 * i + 1] = gcn_w[(k0 + 1) * H_DIM + ncol];
    }
    if (tid < H_DIM) lds_bias[tid] = gcn_b[tid];
    for (int i = tid; i < 3 * B_SZ * H_DIM; i += 256) lds_x[i] = xproj[i];
    __syncthreads();

    int stripe = blockIdx.x * 8 + (tid >> 5);
    if (stripe >= (B_SZ * N_NODES) / 16) return;

    int lane  = tid & 31;
    int m     = lane & 15;       // A-matrix row within stripe
    int khalf = lane >> 4;       // A holds K = k0, k0+1 with k0 = kk*4 + 2*khalf
    int rowBase = stripe * 16;   // global row = b*N + n

    v8f acc[4];
#pragma unroll
    for (int t4 = 0; t4 < 4; ++t4) acc[t4] = (v8f){0.f,0.f,0.f,0.f,0.f,0.f,0.f,0.f};

    const float* arow  = agg + (size_t)(rowBase + m) * H_DIM;
    const v2f*   wlane = (const v2f*)lds_wswz;
#pragma unroll
    for (int kk = 0; kk < 16; ++kk) {
        int k0 = kk * 4 + khalf * 2;
        v2f a = *(const v2f*)(arow + k0);
#pragma unroll
        for (int t4 = 0; t4 < 4; ++t4) {
            v2f bfrag = wlane[(kk * 4 + t4) * 32 + lane];
            acc[t4] = __builtin_amdgcn_wmma_f32_16x16x4_f32(
                false, a, false, bfrag, (short)0, acc[t4], false, false);
        }
    }

    // cooperative re-zero of this stripe's agg tile (16x64 floats, coalesced b128)
    float4* ztile = (float4*)(agg + (size_t)rowBase * H_DIM);
#pragma unroll
    for (int it = 0; it < 8; ++it)
        ztile[it * 32 + lane] = (float4){0.f, 0.f, 0.f, 0.f};

    // epilogue: GRU gates + store h_new into out[:, tstep]
    int b = rowBase / N_NODES;                 // stripes never straddle batches
    int nBase = rowBase - b * N_NODES;
    const float* xr = lds_x + 0 * 128 + b * 64;
    const float* xz = lds_x + 1 * 128 + b * 64;
    const float* xh = lds_x + 2 * 128 + b * 64;
    float* outT = out + ((size_t)(b * SEQ_LEN + tstep)) * N_NODES * H_DIM;
    const float* outPrev = out +
        ((size_t)(b * SEQ_LEN + (tstep > 0 ? tstep - 1 : 0))) * N_NODES * H_DIM;

#pragma unroll
    for (int t4 = 0; t4 < 4; ++t4) {
        int col  = t4 * 16 + (lane & 15);
        float xrv = xr[col], xzv = xz[col], xhv = xh[col];
        float bias = lds_bias[col];
#pragma unroll
        for (int v = 0; v < 8; ++v) {
            int n = nBase + v + khalf * 8;     // C layout: lanes>=16 -> M+8
            float g  = acc[t4][v] + bias;
            float rg = 1.0f / (1.0f + __expf(-(xrv + g)));
            float z  = 1.0f / (1.0f + __expf(-(xzv + g)));
            float ht = tanhf(xhv + rg * g);
            float hold = (tstep == 0) ? 0.0f : outPrev[(size_t)n * H_DIM + col];
            outT[(size_t)n * H_DIM + col] = (1.0f - z) * hold + z * ht;
        }
    }
}

// ---------------------------------------------------------------- launcher
extern "C" void kernel_launch(void* const* d_in, const int* in_sizes, int n_in,
                              void* d_out, int out_size, void* d_ws, size_t ws_size,
                              hipStream_t stream) {
    const float* x     = (const float*)d_in[0];
    const int*   src   = (const int*)  d_in[1];
    const int*   dst   = (const int*)  d_in[2];
    const float* w_r_w = (const float*)d_in[3];
    const float* w_r_b = (const float*)d_in[4];
    const float* w_z_w = (const float*)d_in[5];
    const float* w_z_b = (const float*)d_in[6];
    const float* w_h_w = (const float*)d_in[7];
    const float* w_h_b = (const float*)d_in[8];
    const float* gcn_w = (const float*)d_in[9];
    const float* gcn_b = (const float*)d_in[10];

    float* out    = (float*)d_out;
    float* ws     = (float*)d_ws;
    float* agg    = ws + WS_AGG;
    float* norm_s = ws + WS_NORMS;
    float* norm_d = ws + WS_NORMD;
    float* coeff  = ws + WS_COEFF;
    float* xproj  = ws + WS_XPROJ;

    // per-launch setup — deterministic
    k_zero  <<<(B_SZ * N_NODES * H_DIM + 255) / 256, 256, 0, stream>>>(agg, B_SZ * N_NODES * H_DIM);
    k_zero  <<<(2 * N_NODES + 255) / 256, 256, 0, stream>>>(norm_s, 2 * N_NODES); // norm_s|norm_d contiguous
    k_degree<<<(E_EDGES + 255) / 256, 256, 0, stream>>>(src, dst, norm_s, norm_d);
    k_norm  <<<(2 * N_NODES + 255) / 256, 256, 0, stream>>>(norm_s, 2 * N_NODES);
    k_coeff <<<(E_EDGES + 255) / 256, 256, 0, stream>>>(src, dst, norm_s, norm_d, coeff);
    k_xproj <<<1, 128, 0, stream>>>(x, w_r_w, w_r_b, w_z_w, w_z_b, w_h_w, w_h_b, xproj);

    const int stripes   = (B_SZ * N_NODES) / 16;   // 1250
    const int updBlocks = (stripes + 7) / 8;       // 8 waves per 256-thread block
    const int scatThreads = E_EDGES * 32;

    for (int t = 0; t < SEQ_LEN; ++t) {
        if (t > 0)
            k_scatter<<<(scatThreads + 255) / 256, 256, 0, stream>>>(
                out, t - 1, src, dst, coeff, agg);
        k_update<<<updBlocks, 256, 0, stream>>>(agg, gcn_w, gcn_b, xproj, out, t);
    }
}